// ChessboardLayer_68891275428175
// MI455X (gfx1250) — compile-verified
//
#include <hip/hip_runtime.h>
#include <hip/hip_bf16.h>

// ChessboardLayer: (B=64, H=256, W=256, C=16) f32
//   reshape (B,8,32,8,32,C) -> transpose (B,8,8,32,32,C) -> (B,8,8,16384)
// Pure permutation, 512 MiB traffic -> HBM-bound (~23 us @ 23.3 TB/s).
// Implemented with the CDNA5 Tensor Data Mover: each wave issues one
// 2D-tile tensor_load_to_lds (4 rows x 2KB, row stride 16KB) and one
// contiguous tensor_store_from_lds (8KB), tracked by TENSORcnt.

#define B_DIM 64
#define H_DIM 256
#define W_DIM 256
#define C_DIM 16
#define ROW_F (W_DIM * C_DIM)        // 4096 floats between consecutive h rows
#define CELL_OUT_F (32 * 32 * C_DIM) // 16384 floats per output cell

typedef __attribute__((ext_vector_type(4))) unsigned int u32x4;
typedef __attribute__((ext_vector_type(8))) unsigned int u32x8;

__global__ __launch_bounds__(256)
void ChessboardLayer_68891275428175_kernel(const float* __restrict__ in,
                                           float* __restrict__ out) {
    // One 64 KB cell staged per block; wave w owns the 8 KB slice for
    // rows [4w, 4w+4).
    __shared__ float smem[16384];

    const unsigned cell = blockIdx.x;          // 0..4095
    const int j = cell & 7;
    const int i = (cell >> 3) & 7;
    const int b = cell >> 6;

    // Wave id, forced uniform so all descriptor math stays scalar (SGPRs).
    const int w = __builtin_amdgcn_readfirstlane((int)(threadIdx.x >> 5));

    // Workgroup-relative LDS byte address of this wave's slice. The LDS
    // aperture lives in addr[63:32], so the low 32 bits of the generic
    // pointer are the LDS offset (ISA 10.2).
    const unsigned lds_base =
        __builtin_amdgcn_readfirstlane((unsigned)(unsigned long long)&smem[0])
        + (unsigned)w * 8192u;

    // Global byte address of this wave's input tile: in[b][i*32 + 4w][j*32][0]
    const unsigned long long in_tile =
        (unsigned long long)in +
        (((unsigned long long)(b * H_DIM + i * 32 + w * 4) * W_DIM + j * 32)
         * C_DIM) * 4ull;
    // Global byte address of this wave's output slice: cell base + 4w rows
    const unsigned long long out_tile =
        (unsigned long long)out +
        ((unsigned long long)((b * 8 + i) * 8 + j) * CELL_OUT_F
         + (unsigned)w * 2048u) * 4ull;

    // ---- Tensor DMA descriptor: LOAD (2D tile gather) -------------------
    // Group 0: [1:0]=count=1, [63:32]=lds_addr, [120:64]=global_addr,
    //          [127:126]=type=2
    u32x4 l0;
    l0[0] = 1u;
    l0[1] = lds_base;
    l0[2] = (unsigned)in_tile;
    l0[3] = ((unsigned)(in_tile >> 32) & 0x01FFFFFFu) | 0x80000000u;
    // Group 1: [17:16]=data_size=2 (4B), dim0=512, dim1=4, tile0=512,
    //          tile1=4, dim0_stride=4096 elems; workgroup_mask=0 (no cluster)
    u32x8 l1;
    l1[0] = 2u << 16;          // data_size = 4 bytes
    l1[1] = 512u << 16;        // tensor_dim0[15:0]
    l1[2] = 4u << 16;          // tensor_dim0[31:16]=0 | tensor_dim1[15:0]=4
    l1[3] = 512u << 16;        // tensor_dim1[31:16]=0 | tile_dim0=512
    l1[4] = 4u;                // tile_dim1=4 | tile_dim2=0
    l1[5] = 4096u;             // tensor_dim0_stride[31:0] (elements)
    l1[6] = 0u;                // stride0[47:32]=0 | stride1[15:0]=0
    l1[7] = 0u;

    // ---- Tensor DMA descriptor: STORE (1D contiguous 2048 elems) --------
    u32x4 s0;
    s0[0] = 1u;
    s0[1] = lds_base;
    s0[2] = (unsigned)out_tile;
    s0[3] = ((unsigned)(out_tile >> 32) & 0x01FFFFFFu) | 0x80000000u;
    u32x8 s1;
    s1[0] = 2u << 16;          // data_size = 4 bytes
    s1[1] = 2048u << 16;       // tensor_dim0 = 2048
    s1[2] = 1u << 16;          // tensor_dim1 = 1
    s1[3] = 2048u << 16;       // tile_dim0 = 2048
    s1[4] = 1u;                // tile_dim1 = 1
    s1[5] = 2048u;             // dim0 stride (single row; benign)
    s1[6] = 0u;
    s1[7] = 0u;

    // HBM -> LDS tile gather via TDM (streaming: non-temporal).
    asm volatile("tensor_load_to_lds %0, %1 th:TH_LOAD_NT"
                 :: "s"(l0), "s"(l1) : "memory");
    // Load must fully land in LDS before the store descriptor reads it.
    __builtin_amdgcn_s_wait_tensorcnt(0);
    // LDS -> HBM contiguous write via TDM.
    asm volatile("tensor_store_from_lds %0, %1 th:TH_STORE_NT"
                 :: "s"(s0), "s"(s1) : "memory");
    // S_ENDPGM performs an implicit wait-idle; the tensor store drains.
}

extern "C" void kernel_launch(void* const* d_in, const int* in_sizes, int n_in,
                              void* d_out, int out_size, void* d_ws, size_t ws_size,
                              hipStream_t stream) {
    (void)in_sizes; (void)n_in; (void)d_ws; (void)ws_size; (void)out_size;
    const float* in = (const float*)d_in[0];
    float* out = (float*)d_out;
    // One block per (b, i, j) cell: 64*8*8 = 4096 blocks, 8 waves each.
    dim3 grid(B_DIM * 8 * 8), block(256);
    ChessboardLayer_68891275428175_kernel<<<grid, block, 0, stream>>>(in, out);
}